// FilterDetections_45543833207114
// MI455X (gfx1250) — compile-verified
//
#include <hip/hip_runtime.h>

typedef unsigned long long u64;
typedef unsigned int u32;

// ---- problem constants (match reference setup_inputs) ----
static constexpr int   B      = 2;
static constexpr int   N      = 100000;
static constexpr int   C      = 80;
static constexpr int   K      = 300;       // MAX_DETECTIONS
static constexpr int   NB     = 128;       // score histogram bins over (0,1]
static constexpr int   CAP    = 2048;      // per-class candidate cap (pow2)
static constexpr float THR    = 0.05f;     // SCORE_THRESHOLD
static constexpr float NMST   = 0.5f;      // NMS_THRESHOLD
static constexpr int   CHUNK  = 80000;     // elems/WG in streaming kernels (div by 80 and 4)
static constexpr int   NCHUNK = (N * C) / CHUNK;  // 100
static constexpr int   FLATC  = C * K;     // 24000 candidates per image
static constexpr int   SEG    = 4096;      // final top-k segment sort size (pow2)
static constexpr int   NSEG   = 6;         // 6*4096 = 24576 >= 24000
static constexpr int   KW     = (K + 31) / 32;    // 10 bitmask words per row
static constexpr int   NT     = (K + 15) / 16;    // 19 tiles of 16 along each axis

// ---- CDNA5 async global->LDS path (probe-confirmed present on this toolchain) ----
#if defined(__has_builtin)
#if __has_builtin(__builtin_amdgcn_global_load_async_to_lds_b128) && \
    __has_builtin(__builtin_amdgcn_s_wait_asynccnt)
#define USE_ASYNC_LDS 1
#endif
#if __has_builtin(__builtin_amdgcn_wmma_f32_16x16x4_f32)
#define USE_WMMA 1
#endif
#endif

#if USE_ASYNC_LDS
// builtin expects a pointer to a 16-byte int vector (per hipcc diagnostic)
typedef int v4i __attribute__((ext_vector_type(4)));
typedef __attribute__((address_space(1))) v4i* gv4p;  // global src
typedef __attribute__((address_space(3))) v4i* lv4p;  // LDS dst
#endif

#if USE_WMMA
typedef float v2f __attribute__((ext_vector_type(2)));
typedef float v8f __attribute__((ext_vector_type(8)));
#endif

#define DEVFN __device__ __forceinline__

// monotone float<->uint mapping (descending u64-key sorts give descending scores)
DEVFN u32 ordf(float s) {
  u32 u = __float_as_uint(s);
  return (u & 0x80000000u) ? ~u : (u | 0x80000000u);
}
DEVFN float unordf(u32 ou) {
  return __uint_as_float((ou & 0x80000000u) ? (ou ^ 0x80000000u) : ~ou);
}

// in-LDS bitonic sort, descending, M = power of two, called by all threads
DEVFN void bitonic_desc(u64* a, int M, int tid, int T) {
  for (int k = 2; k <= M; k <<= 1) {
    for (int j = k >> 1; j > 0; j >>= 1) {
      for (int i = tid; i < M; i += T) {
        int ixj = i ^ j;
        if (ixj > i) {
          u64 x = a[i], y = a[ixj];
          bool desc = ((i & k) == 0);
          if (desc ? (x < y) : (x > y)) { a[i] = y; a[ixj] = x; }
        }
      }
      __syncthreads();
    }
  }
}

// ---------------- kernel 0: zero histograms + counters ----------------
__global__ void k_zero(int* __restrict__ hist, int* __restrict__ gcount) {
  int i = blockIdx.x * blockDim.x + threadIdx.x;
  if (i < B * C * NB) hist[i] = 0;
  if (i < B * C) gcount[i] = 0;
}

// ---------------- kernel 1: per-class score histograms ----------------
// LDS-privatized 80x128 histogram per WG (40KB of the 320KB WGP LDS);
// scores streamed with double-buffered async global->LDS staging (ASYNCcnt).
__global__ __launch_bounds__(256) void k_hist(const float* __restrict__ cls,
                                              int* __restrict__ hist) {
  __shared__ int lh[C * NB];   // 40 KB
  const int tid   = threadIdx.x;
  const int b     = blockIdx.x / NCHUNK;
  const int chunk = blockIdx.x % NCHUNK;
  const long long g0 = (long long)b * (N * C) + (long long)chunk * CHUNK;

  for (int i = tid; i < C * NB; i += 256) lh[i] = 0;
  __syncthreads();

#if USE_ASYNC_LDS
  __shared__ float stage[2][1024];  // 2 x 4KB double buffer
  const int R = (CHUNK + 1023) / 1024;  // rounds of 1024 floats
  {
    int idx = tid * 4;
    int idc = (idx < CHUNK) ? idx : 0;  // clamp: keep per-wave ASYNCcnt uniform
    __builtin_amdgcn_global_load_async_to_lds_b128(
        (gv4p)(cls + g0 + idc), (lv4p)(&stage[0][tid * 4]), 0, 0);
  }
  for (int r = 0; r < R; ++r) {
    if (r + 1 < R) {
      int idx = (r + 1) * 1024 + tid * 4;
      int idc = (idx < CHUNK) ? idx : 0;
      __builtin_amdgcn_global_load_async_to_lds_b128(
          (gv4p)(cls + g0 + idc), (lv4p)(&stage[(r + 1) & 1][tid * 4]), 0, 0);
      __builtin_amdgcn_s_wait_asynccnt(1);  // round r landed (in-order completion)
    } else {
      __builtin_amdgcn_s_wait_asynccnt(0);
    }
    __syncthreads();
    int idx = r * 1024 + tid * 4;
    if (idx < CHUNK) {
      float4 v = *(const float4*)(&stage[r & 1][tid * 4]);
      float vv[4] = {v.x, v.y, v.z, v.w};
#pragma unroll
      for (int j = 0; j < 4; ++j) {
        float s = vv[j];
        if (s > THR) {
          int c = (idx + j) % C;  // chunk base divisible by 80
          int bin = (int)(s * (float)NB);
          if (bin > NB - 1) bin = NB - 1;
          atomicAdd(&lh[c * NB + bin], 1);
        }
      }
    }
    __syncthreads();  // done reading buffer before it is refilled
  }
#else
  for (int i = tid * 4; i < CHUNK; i += 256 * 4) {
    float4 v = *(const float4*)(cls + g0 + i);
    float vv[4] = {v.x, v.y, v.z, v.w};
#pragma unroll
    for (int j = 0; j < 4; ++j) {
      float s = vv[j];
      if (s > THR) {
        int c = (i + j) % C;
        int bin = (int)(s * (float)NB);
        if (bin > NB - 1) bin = NB - 1;
        atomicAdd(&lh[c * NB + bin], 1);
      }
    }
  }
#endif
  __syncthreads();
  for (int i = tid; i < C * NB; i += 256) {
    int v = lh[i];
    if (v) atomicAdd(&hist[b * C * NB + i], v);
  }
}

// ---------------- kernel 2: cutoff bin containing the K-th score ----------------
__global__ void k_cut(const int* __restrict__ hist, int* __restrict__ cutbin) {
  int t = threadIdx.x;
  if (t < B * C) {
    const int* h = hist + t * NB;
    int acc = 0, cb = 0;
    for (int bin = NB - 1; bin >= 0; --bin) {
      acc += h[bin];
      if (acc >= K) { cb = bin; break; }
    }
    cutbin[t] = cb;  // acc<K overall -> cb=0 -> collect everything > THR
  }
}

// ---------------- kernel 3: collect candidates >= cutoff (L2-resident pass) ----------------
__global__ __launch_bounds__(256) void k_collect(const float* __restrict__ cls,
                                                 const int* __restrict__ cutbin,
                                                 int* __restrict__ gcount,
                                                 u64* __restrict__ gcand) {
  __shared__ int scb[B * C];
  const int tid = threadIdx.x;
  for (int i = tid; i < B * C; i += 256) scb[i] = cutbin[i];
  __syncthreads();
  const int b     = blockIdx.x / NCHUNK;
  const int chunk = blockIdx.x % NCHUNK;
  const long long base = (long long)chunk * CHUNK;
  const float* p = cls + (long long)b * (N * C);
  for (int i = tid * 4; i < CHUNK; i += 256 * 4) {
    if (base + i + 8192 < (long long)N * C)
      __builtin_prefetch(p + base + i + 8192, 0, 0);  // global_prefetch_b8
    float4 v = *(const float4*)(p + base + i);
    float vv[4] = {v.x, v.y, v.z, v.w};
#pragma unroll
    for (int j = 0; j < 4; ++j) {
      float s = vv[j];
      if (s > THR) {
        long long off = base + i + j;
        int c = (int)(off % C);
        int bin = (int)(s * (float)NB);
        if (bin > NB - 1) bin = NB - 1;
        if (bin >= scb[b * C + c]) {
          int pos = atomicAdd(&gcount[b * C + c], 1);
          if (pos < CAP) {
            u32 n = (u32)(off / C);  // anchor index
            gcand[(size_t)(b * C + c) * CAP + pos] = ((u64)ordf(s) << 32) | (u64)n;
          }
        }
      }
    }
  }
}

// ---------------- kernel 4: per-(image,class) sort -> top-300 -> NMS ----------------
// Suppression matrix built in parallel: the all-pairs union base (area_i + area_j)
// comes from V_WMMA_F32_16X16X4_F32 tiles (exact: products are area*1, 1*area),
// intersection from per-lane min/max; greedy scan is a serial bitmask sweep.
__global__ __launch_bounds__(512) void k_sortnms(const u64* __restrict__ gcand,
                                                 const int* __restrict__ gcount,
                                                 const float* __restrict__ boxes,
                                                 u64* __restrict__ flat_keys,
                                                 float* __restrict__ cand_boxes) {
  __shared__ u64 sk[CAP];                                  // 16 KB
  __shared__ float sy1[K], sx1[K], sy2[K], sx2[K], sar[K]; // 6 KB
  __shared__ u32 supT[K * KW];                             // 12 KB: sup bits, j<i only
  __shared__ int kept[K], vld[K];
  const int tid = threadIdx.x;
  const int bc = blockIdx.x;
  const int b = bc / C, c = bc % C;

  int cnt = gcount[bc];
  if (cnt > CAP) cnt = CAP;
  for (int i = tid; i < CAP; i += 512)
    sk[i] = (i < cnt) ? gcand[(size_t)bc * CAP + i] : 0ull;
  for (int i = tid; i < K * KW; i += 512) supT[i] = 0u;
  __syncthreads();
  bitonic_desc(sk, CAP, tid, 512);

  if (tid < K) {
    u64 key = sk[tid];
    u32 n = (u32)(key & 0xffffffffull);
    const float* bp = boxes + ((size_t)b * N + n) * 4;
    float y1 = bp[0], x1 = bp[1], y2 = bp[2], x2 = bp[3];
    sy1[tid] = y1; sx1[tid] = x1; sy2[tid] = y2; sx2[tid] = x2;
    sar[tid] = (y2 - y1) * (x2 - x1);
    vld[tid] = (key != 0ull);
  }
  __syncthreads();

  // ---- build lower-triangular suppression bitmask, one 16x16 pair tile per wave ----
  const int lane = tid & 31;
  const int wid = tid >> 5;                 // 16 waves
  for (int t = wid; t < NT * NT; t += 16) {
    const int ti = t / NT, tj = t % NT;
    if (tj > ti) continue;                  // wave-uniform: EXEC stays all-ones
#if USE_WMMA
    // A (16x4): col0 = area_i, col1 = 1 ; B (4x16): row0 = 1, row1 = area_j
    int ai = ti * 16 + lane, bj = tj * 16 + lane;
    v2f av, bv;
    av.x = (lane < 16) ? ((ai < K) ? sar[ai] : 0.0f) : 0.0f;  // K=0 / K=2 halves
    av.y = (lane < 16) ? 1.0f : 0.0f;                          // K=1 / K=3 halves
    bv.x = (lane < 16) ? 1.0f : 0.0f;
    bv.y = (lane < 16) ? ((bj < K) ? sar[bj] : 0.0f) : 0.0f;
    v8f acc = {};
    acc = __builtin_amdgcn_wmma_f32_16x16x4_f32(
        false, av, false, bv, (short)0, acc, false, false);  // D[m][n]=area_i+area_j
#endif
#pragma unroll
    for (int r = 0; r < 8; ++r) {
      int m = r + ((lane >> 4) << 3);       // C/D layout: VGPR r -> M=r / M=r+8
      int n = lane & 15;
      int gi = ti * 16 + m, gj = tj * 16 + n;
      if (gi < K && gj < gi) {
        float ty = fmaxf(sy1[gi], sy1[gj]);
        float tx = fmaxf(sx1[gi], sx1[gj]);
        float by = fminf(sy2[gi], sy2[gj]);
        float bx = fminf(sx2[gi], sx2[gj]);
        float inter = fmaxf(by - ty, 0.0f) * fmaxf(bx - tx, 0.0f);
#if USE_WMMA
        float areasum = acc[r];
#else
        float areasum = sar[gi] + sar[gj];
#endif
        float uni = areasum - inter;
        // iou > 0.5  <=>  inter > 0.5 * max(union, 1e-8): avoids the full IEEE
        // f32 division sequence (v_div_scale/v_rcp/v_div_fmas) in the hot loop
        if (inter > NMST * fmaxf(uni, 1e-8f))
          atomicOr(&supT[gi * KW + (gj >> 5)], 1u << (gj & 31));
      }
    }
  }
  __syncthreads();

  // ---- greedy scan: serial bitmask sweep (no barriers) ----
  if (tid == 0) {
    u32 kw[KW];
#pragma unroll
    for (int w = 0; w < KW; ++w) kw[w] = 0u;
    for (int i = 0; i < K; ++i) {
      u32 any = 0;
#pragma unroll
      for (int w = 0; w < KW; ++w) any |= kw[w] & supT[i * KW + w];
      int keep = vld[i] && (any == 0u);
      kept[i] = keep;
      if (keep) kw[i >> 5] |= 1u << (i & 31);
    }
  }
  __syncthreads();

  if (tid < K) {
    u64 key = sk[tid];
    int p = c * K + tid;                  // flat (class,rank) index within image
    size_t fo = (size_t)b * FLATC + p;
    flat_keys[fo] = kept[tid] ? ((key & 0xffffffff00000000ull) | (u64)(u32)p) : 0ull;
    float* cb = cand_boxes + fo * 4;
    cb[0] = sy1[tid]; cb[1] = sx1[tid]; cb[2] = sy2[tid]; cb[3] = sx2[tid];
  }
}

// ---------------- kernel 5: per-image global top-300 + output ----------------
__global__ __launch_bounds__(1024) void k_final(const u64* __restrict__ flat_keys,
                                                const float* __restrict__ cand_boxes,
                                                float* __restrict__ out) {
  __shared__ u64 sk[SEG];          // 32 KB
  __shared__ u64 stop[NSEG * K];   // 14.4 KB: segment-wise top-300s
  const int tid = threadIdx.x;
  const int b = blockIdx.x;

  for (int seg = 0; seg < NSEG; ++seg) {
    for (int i = tid; i < SEG; i += 1024) {
      int g = seg * SEG + i;
      sk[i] = (g < FLATC) ? flat_keys[(size_t)b * FLATC + g] : 0ull;
    }
    __syncthreads();
    bitonic_desc(sk, SEG, tid, 1024);
    if (tid < K) stop[seg * K + tid] = sk[tid];
    __syncthreads();
  }
  // exact: overall top-300 is contained in the union of segment top-300s
  for (int i = tid; i < 2048; i += 1024)
    sk[i] = (i < NSEG * K) ? stop[i] : 0ull;
  __syncthreads();
  bitonic_desc(sk, 2048, tid, 1024);

  if (tid < K) {
    u64 key = sk[tid];
    float b0, b1, b2, b3, sc; int lab;
    if (key != 0ull) {
      u32 p = (u32)(key & 0xffffffffull);
      const float* cb = cand_boxes + ((size_t)b * FLATC + p) * 4;
      b0 = cb[0]; b1 = cb[1]; b2 = cb[2]; b3 = cb[3];
      sc = unordf((u32)(key >> 32));
      lab = (int)(p / K);
    } else {
      b0 = b1 = b2 = b3 = -1.0f; sc = -1.0f; lab = -1;
    }
    // d_out = [boxes (B,300,4)][scores (B,300)][labels (B,300) as int32]
    float* ob = out + ((size_t)b * K + tid) * 4;
    ob[0] = b0; ob[1] = b1; ob[2] = b2; ob[3] = b3;
    out[(size_t)B * K * 4 + b * K + tid] = sc;
    ((int*)out)[B * K * 4 + B * K + b * K + tid] = lab;
  }
}

extern "C" void kernel_launch(void* const* d_in, const int* in_sizes, int n_in,
                              void* d_out, int out_size, void* d_ws, size_t ws_size,
                              hipStream_t stream) {
  (void)in_sizes; (void)n_in; (void)out_size; (void)ws_size;
  const float* boxes = (const float*)d_in[0];   // [B,N,4]
  const float* cls   = (const float*)d_in[1];   // [B,N,C]
  float* out = (float*)d_out;

  char* ws = (char*)d_ws;
  int* hist        = (int*)(ws);                 //  81,920 B
  int* cutbin      = (int*)(ws + 81920);         //     640 B
  int* gcount      = (int*)(ws + 82560);         //     640 B
  u64* gcand       = (u64*)(ws + 83200);         // 2,621,440 B
  u64* flat_keys   = (u64*)(ws + 2704640);       //   384,000 B
  float* cand_boxes = (float*)(ws + 3088640);    //   768,000 B  (total ~3.86 MB)

  k_zero   <<<80, 256, 0, stream>>>(hist, gcount);
  k_hist   <<<B * NCHUNK, 256, 0, stream>>>(cls, hist);
  k_cut    <<<1, 256, 0, stream>>>(hist, cutbin);
  k_collect<<<B * NCHUNK, 256, 0, stream>>>(cls, cutbin, gcount, gcand);
  k_sortnms<<<B * C, 512, 0, stream>>>(gcand, gcount, boxes, flat_keys, cand_boxes);
  k_final  <<<B, 1024, 0, stream>>>(flat_keys, cand_boxes, out);
}